// Net_20813411516894
// MI455X (gfx1250) — compile-verified
//
#include <hip/hip_runtime.h>
#include <hip/hip_bf16.h>
#include <math.h>

#define NN    100000
#define EE    3200000
#define FIN   256
#define FHID  16
#define FOUT  40
#define NTILES (NN / 16)   // 6250, N divides 16 exactly

typedef __attribute__((ext_vector_type(2))) float v2f;
typedef __attribute__((ext_vector_type(8))) float v8f;

// ---------------- degree / norm ----------------
__global__ void k_deg_init(float* __restrict__ deg) {
    int n = blockIdx.x * blockDim.x + threadIdx.x;
    if (n < NN) deg[n] = 1.0f;   // self-loop weight
}

__global__ void k_deg_edges(const int* __restrict__ ei, const float* __restrict__ ew,
                            float* __restrict__ deg) {
    int e = blockIdx.x * blockDim.x + threadIdx.x;
    if (e < EE) atomicAdd(&deg[ei[EE + e]], ew[e]);   // col = ei[1][e]
}

__global__ void k_dinv(float* __restrict__ deg) {
    int n = blockIdx.x * blockDim.x + threadIdx.x;
    if (n < NN) {
        float d = deg[n];
        deg[n] = (d > 0.0f) ? rsqrtf(d) : 0.0f;       // in-place -> dinv
    }
}

// ---------------- layer-1 GEMM: h = x @ W1, agg1 = h * dinv^2 (self loop) ----------------
// One wave computes a 16x16 output tile via 64x V_WMMA_F32_16X16X4_F32.
// A (16x4 f32): lane m=lane&15, half=lane>>4 -> VGPR0=x[row][k+2*half], VGPR1=x[row][k+2*half+1]
// B (4x16 f32): lane n=lane&15           -> VGPR0=W1[k+2*half][n],   VGPR1=W1[k+2*half+1][n]
// C/D (16x16 f32): VGPR i -> row (i + 8*half), col n.
__global__ void __launch_bounds__(256) k_gemm1(const float* __restrict__ x,
                                               const float* __restrict__ W1,
                                               const float* __restrict__ dinv,
                                               float* __restrict__ h,
                                               float* __restrict__ agg1) {
    __shared__ float sW1[FIN * FHID];   // 16 KB of 320 KB/WGP LDS
    for (int i = threadIdx.x; i < FIN * FHID; i += blockDim.x) sW1[i] = W1[i];
    __syncthreads();

    const int wave = threadIdx.x >> 5;
    const int lane = threadIdx.x & 31;
    const int tile = blockIdx.x * 8 + wave;
    if (tile >= NTILES) return;                 // wave-uniform: EXEC stays all-1s for WMMA

    const int row0  = tile * 16;
    const int half  = lane >> 4;                // 0|1
    const int idx16 = lane & 15;                // m for A, n for B/C/D

    const float* __restrict__ xrow = x + (size_t)(row0 + idx16) * FIN;

    v8f c = {0.f, 0.f, 0.f, 0.f, 0.f, 0.f, 0.f, 0.f};
    #pragma unroll 8
    for (int k = 0; k < FIN; k += 4) {
        v2f a, b;
        a.x = xrow[k + 2 * half];
        a.y = xrow[k + 2 * half + 1];
        b.x = sW1[(k + 2 * half) * FHID + idx16];
        b.y = sW1[(k + 2 * half + 1) * FHID + idx16];
        c = __builtin_amdgcn_wmma_f32_16x16x4_f32(
                /*neg_a=*/false, a, /*neg_b=*/false, b,
                /*c_mod=*/(short)0, c, /*reuse_a=*/false, /*reuse_b=*/false);
    }

    #pragma unroll
    for (int i = 0; i < 8; ++i) {
        int   row = row0 + i + 8 * half;
        float di  = dinv[row];
        float v   = c[i];
        h[(size_t)row * FHID + idx16]    = v;
        agg1[(size_t)row * FHID + idx16] = v * di * di;   // self-loop: norm = dinv^2
    }
}

// ---------------- edge scatter, layer 1 (16 feats) ----------------
__global__ void k_scatter1(const int* __restrict__ ei, const float* __restrict__ ew,
                           const float* __restrict__ dinv, const float* __restrict__ h,
                           float* __restrict__ agg1) {
    int e = blockIdx.x * blockDim.x + threadIdx.x;
    if (e >= EE) return;
    int r = ei[e];
    int c = ei[EE + e];
    float norm = dinv[r] * ew[e] * dinv[c];
    const float4* __restrict__ hr = (const float4*)(h + (size_t)r * FHID);
    float* __restrict__ dst = agg1 + (size_t)c * FHID;
    #pragma unroll
    for (int q = 0; q < 4; ++q) {
        float4 v = hr[q];
        atomicAdd(dst + 4 * q + 0, v.x * norm);
        atomicAdd(dst + 4 * q + 1, v.y * norm);
        atomicAdd(dst + 4 * q + 2, v.z * norm);
        atomicAdd(dst + 4 * q + 3, v.w * norm);
    }
}

// ---------------- layer 2: h1=relu(agg1+b1); h2=h1@W2; out=b2+h2*dinv^2 ----------------
__global__ void __launch_bounds__(256) k_layer2(const float* __restrict__ agg1,
                                                const float* __restrict__ b1,
                                                const float* __restrict__ W2,
                                                const float* __restrict__ b2,
                                                const float* __restrict__ dinv,
                                                float* __restrict__ h2,
                                                float* __restrict__ out) {
    __shared__ float sW2[FHID * FOUT];
    __shared__ float sb1[FHID];
    __shared__ float sb2[FOUT];
    for (int i = threadIdx.x; i < FHID * FOUT; i += blockDim.x) sW2[i] = W2[i];
    if (threadIdx.x < FHID) sb1[threadIdx.x] = b1[threadIdx.x];
    if (threadIdx.x < FOUT) sb2[threadIdx.x] = b2[threadIdx.x];
    __syncthreads();

    int n = blockIdx.x * blockDim.x + threadIdx.x;
    if (n >= NN) return;

    float h1[FHID];
    const float* __restrict__ a = agg1 + (size_t)n * FHID;
    #pragma unroll
    for (int k = 0; k < FHID; ++k) {
        float v = a[k] + sb1[k];
        h1[k] = v > 0.0f ? v : 0.0f;
    }
    float di = dinv[n];
    float d2 = di * di;
    float* __restrict__ hrow = h2 + (size_t)n * FOUT;
    float* __restrict__ orow = out + (size_t)n * FOUT;
    #pragma unroll 4
    for (int j = 0; j < FOUT; ++j) {
        float acc = 0.0f;
        #pragma unroll
        for (int k = 0; k < FHID; ++k) acc = fmaf(h1[k], sW2[k * FOUT + j], acc);
        hrow[j] = acc;
        orow[j] = sb2[j] + acc * d2;      // bias + self-loop contribution
    }
}

// ---------------- edge scatter, layer 2 (40 feats) ----------------
__global__ void k_scatter2(const int* __restrict__ ei, const float* __restrict__ ew,
                           const float* __restrict__ dinv, const float* __restrict__ h2,
                           float* __restrict__ out) {
    int e = blockIdx.x * blockDim.x + threadIdx.x;
    if (e >= EE) return;
    int r = ei[e];
    int c = ei[EE + e];
    float norm = dinv[r] * ew[e] * dinv[c];
    const float4* __restrict__ hr = (const float4*)(h2 + (size_t)r * FOUT);  // 160B, 16B-aligned
    float* __restrict__ dst = out + (size_t)c * FOUT;
    #pragma unroll
    for (int q = 0; q < 10; ++q) {
        float4 v = hr[q];
        atomicAdd(dst + 4 * q + 0, v.x * norm);
        atomicAdd(dst + 4 * q + 1, v.y * norm);
        atomicAdd(dst + 4 * q + 2, v.z * norm);
        atomicAdd(dst + 4 * q + 3, v.w * norm);
    }
}

// ---------------- in-place log_softmax over 40 classes ----------------
__global__ void k_logsoftmax(float* __restrict__ out) {
    int n = blockIdx.x * blockDim.x + threadIdx.x;
    if (n >= NN) return;
    float* __restrict__ r = out + (size_t)n * FOUT;
    float v[FOUT];
    float m = -INFINITY;
    #pragma unroll
    for (int j = 0; j < FOUT; ++j) { v[j] = r[j]; m = fmaxf(m, v[j]); }
    float s = 0.0f;
    #pragma unroll
    for (int j = 0; j < FOUT; ++j) s += __expf(v[j] - m);
    float lse = __logf(s) + m;
    #pragma unroll
    for (int j = 0; j < FOUT; ++j) r[j] = v[j] - lse;
}

extern "C" void kernel_launch(void* const* d_in, const int* in_sizes, int n_in,
                              void* d_out, int out_size, void* d_ws, size_t ws_size,
                              hipStream_t stream) {
    const float* x  = (const float*)d_in[0];   // [N, 256]
    const int*   ei = (const int*)  d_in[1];   // [2, E] (int32; JAX x64-off demotes int64)
    const float* ew = (const float*)d_in[2];   // [E]
    const float* W1 = (const float*)d_in[3];   // [256, 16]
    const float* b1 = (const float*)d_in[4];   // [16]
    const float* W2 = (const float*)d_in[5];   // [16, 40]
    const float* b2 = (const float*)d_in[6];   // [40]
    float* out = (float*)d_out;                // [N, 40]

    // workspace: deg/dinv | h | agg1 | h2   (~29.2 MB)
    float* deg  = (float*)d_ws;
    float* h    = deg  + NN;
    float* agg1 = h    + (size_t)NN * FHID;
    float* h2   = agg1 + (size_t)NN * FHID;

    const int TB = 256;
    const int gN = (NN + TB - 1) / TB;         // 391
    const int gE = (EE + TB - 1) / TB;         // 12500
    const int gT = (NTILES + 7) / 8;           // 782 blocks, 8 waves (tiles) each

    k_deg_init  <<<gN, TB, 0, stream>>>(deg);
    k_deg_edges <<<gE, TB, 0, stream>>>(ei, ew, deg);
    k_dinv      <<<gN, TB, 0, stream>>>(deg);                    // deg -> dinv in place
    k_gemm1     <<<gT, TB, 0, stream>>>(x, W1, deg, h, agg1);    // WMMA f32
    k_scatter1  <<<gE, TB, 0, stream>>>(ei, ew, deg, h, agg1);
    k_layer2    <<<gN, TB, 0, stream>>>(agg1, b1, W2, b2, deg, h2, out);
    k_scatter2  <<<gE, TB, 0, stream>>>(ei, ew, deg, h2, out);
    k_logsoftmax<<<gN, TB, 0, stream>>>(out);
}